// DotProductPredictor_56307021251124
// MI455X (gfx1250) — compile-verified
//
#include <hip/hip_runtime.h>
#include <hip/hip_bf16.h>

typedef __attribute__((ext_vector_type(2))) float v2f;
typedef __attribute__((ext_vector_type(8))) float v8f;

#define D_FEAT 512
#define WAVES_PER_BLOCK 8
#define TILE_EDGES 16

// One wave32 handles one tile of 16 edges.
// A-tile: 16 rows of h[src[...]], B-tile: 16 rows of h[dst[...]] (as K x N).
// Accumulate C = A*B^T via v_wmma_f32_16x16x4_f32 over K=512 in steps of 4,
// then extract the 16 diagonal elements (the per-edge dot products).
__global__ __launch_bounds__(WAVES_PER_BLOCK * 32, 1)
void sddmm_udotv_wmma_f32(const float* __restrict__ h,
                          const int* __restrict__ src,
                          const int* __restrict__ dst,
                          float* __restrict__ out,
                          int nEdges, long nTiles) {
    const int lane = (int)(threadIdx.x & 31u);
    const int wave = (int)(threadIdx.x >> 5u);
    const long tile = (long)blockIdx.x * WAVES_PER_BLOCK + wave;
    if (tile >= nTiles) return;   // wave-uniform: EXEC stays all-1s below (WMMA requirement)

    const long edgeBase = tile * TILE_EDGES;
    const int  m        = lane & 15;            // tile row this lane feeds
    long eIdx = edgeBase + m;
    if (eIdx >= (long)nEdges) eIdx = (long)nEdges - 1;  // clamp; result masked at store

    const int si = src[eIdx];
    const int di = dst[eIdx];

    // ISA 16x4 f32 A layout: lanes 0-15 hold row m, K = k..k+1 (2 VGPRs);
    // lanes 16-31 hold row m, K = k+2..k+3. B (4x16, B[k][n]=h[dst[n]][k])
    // has the mirrored layout -> identical per-lane addressing.
    const int colOff = (lane & 16) ? 2 : 0;
    const float* __restrict__ aRow = h + (long)si * D_FEAT + colOff;
    const float* __restrict__ bRow = h + (long)di * D_FEAT + colOff;

    v8f acc = {};
#pragma unroll 8
    for (int k = 0; k < D_FEAT; k += 4) {
        v2f a = *(const v2f*)(aRow + k);   // global_load_b64
        v2f b = *(const v2f*)(bRow + k);   // global_load_b64
        // D = A x B + C   (8 args: neg_a, A, neg_b, B, c_mod, C, reuse_a, reuse_b)
        acc = __builtin_amdgcn_wmma_f32_16x16x4_f32(
            false, a, false, b, (short)0, acc, false, false);
    }

    // Diagonal extraction from C/D layout:
    //   m in 0..7  -> VGPR m,   lane m
    //   m in 8..15 -> VGPR m-8, lane m+16
    const int r = lane & 7;
    float v = acc[0];
    v = (r == 1) ? acc[1] : v;
    v = (r == 2) ? acc[2] : v;
    v = (r == 3) ? acc[3] : v;
    v = (r == 4) ? acc[4] : v;
    v = (r == 5) ? acc[5] : v;
    v = (r == 6) ? acc[6] : v;
    v = (r == 7) ? acc[7] : v;

    const bool writer = (lane < 8) || (lane >= 24);
    const int  mo     = (lane < 8) ? lane : (lane - 16);
    const long oIdx   = edgeBase + mo;
    if (writer && oIdx < (long)nEdges) {
        __builtin_nontemporal_store(v, &out[oIdx]);  // don't let the 6.4MB output evict h from L2
    }
}

extern "C" void kernel_launch(void* const* d_in, const int* in_sizes, int n_in,
                              void* d_out, int out_size, void* d_ws, size_t ws_size,
                              hipStream_t stream) {
    const float* h   = (const float*)d_in[0];
    const int*   src = (const int*)d_in[1];
    const int*   dst = (const int*)d_in[2];
    float*       out = (float*)d_out;

    const int  nEdges = in_sizes[1];
    const long nTiles = ((long)nEdges + TILE_EDGES - 1) / TILE_EDGES;
    const int  blocks = (int)((nTiles + WAVES_PER_BLOCK - 1) / WAVES_PER_BLOCK);

    sddmm_udotv_wmma_f32<<<blocks, WAVES_PER_BLOCK * 32, 0, stream>>>(
        h, src, dst, out, nEdges, nTiles);
}